// ProductManifoldGNN_59631325937739
// MI455X (gfx1250) — compile-verified
//
#include <hip/hip_runtime.h>
#include <math.h>

typedef __attribute__((ext_vector_type(2))) float v2f;
typedef __attribute__((ext_vector_type(8))) float v8f;

#define RCUT 3.5f
#define PI_F 3.14159265358979323846f
#define S2C 0.70710678118654752f
#define S6C 0.40824829046386302f
#define SQRT3C 1.73205080756887729f
#define SQRT15C 3.87298334620741688f

static __device__ __forceinline__ v8f wmma4(v2f a, v2f b, v8f c) {
    // D = A(16x4 f32) x B(4x16 f32) + C(16x16 f32)
    return __builtin_amdgcn_wmma_f32_16x16x4_f32(false, a, false, b, (short)0, c,
                                                 false, false);
}

static __device__ __forceinline__ float siluf(float x) {
    return x / (1.0f + expf(-x));
}
static __device__ __forceinline__ float sigf(float x) {
    return 1.0f / (1.0f + expf(-x));
}
static __device__ __forceinline__ float fin0(float x) {
    return isfinite(x) ? x : 0.0f;
}

// ---------------------------------------------------------------------------
// Stage 1: per-edge geometry + radial MLP + edge attrs + neighbor-embed scatter
// ---------------------------------------------------------------------------
__global__ __launch_bounds__(256)
void pmg_edge_pre_kernel(const float* __restrict__ pos,
                         const int* __restrict__ ei,      // [2,E]
                         const float* __restrict__ means,
                         const float* __restrict__ betas,
                         const float* __restrict__ ne_dw,  // [8,32]
                         const float* __restrict__ ne_db,  // [32]
                         const float* __restrict__ ne_emb, // [100,32]
                         const int* __restrict__ z,
                         const float* __restrict__ rad_w1, // [3,8,256]
                         const float* __restrict__ rad_b1, // [3,256]
                         const float* __restrict__ rad_w2, // [3,256]
                         const float* __restrict__ rad_b2, // [3]
                         float* __restrict__ e0g,          // [E]
                         float* __restrict__ e1g,          // [E,3]
                         float* __restrict__ qeg,          // [E,9]
                         float* __restrict__ aggr,         // [N,32]
                         int E) {
    __shared__ float w1s[3 * 8 * 256];
    __shared__ float b1s[3 * 256];
    __shared__ float w2s[3 * 256];
    for (int i = threadIdx.x; i < 3 * 8 * 256; i += blockDim.x) w1s[i] = rad_w1[i];
    for (int i = threadIdx.x; i < 3 * 256; i += blockDim.x) {
        b1s[i] = rad_b1[i];
        w2s[i] = rad_w2[i];
    }
    __syncthreads();

    int e = blockIdx.x * blockDim.x + threadIdx.x;
    if (e >= E) return;
    int sn = ei[e];
    int dn = ei[E + e];
    float rx = pos[dn * 3 + 0] - pos[sn * 3 + 0];
    float ry = pos[dn * 3 + 1] - pos[sn * 3 + 1];
    float rz = pos[dn * 3 + 2] - pos[sn * 3 + 2];
    float d = sqrtf(rx * rx + ry * ry + rz * rz + 1e-12f);
    float inv = 1.0f / d;
    float hx = rx * inv, hy = ry * inv, hz = rz * inv;

    float C = (d < RCUT) ? 0.5f * (cosf(d * PI_F / RCUT) + 1.0f) : 0.0f;
    float dc = fminf(fmaxf(d, 1e-6f), RCUT);
    float Cc = (dc < RCUT) ? 0.5f * (cosf(dc * PI_F / RCUT) + 1.0f) : 0.0f;
    float ex = expf(-(5.0f / RCUT) * dc);

    float rbf[8];
#pragma unroll
    for (int k = 0; k < 8; ++k) {
        float t = ex - means[k];
        float b = fmaxf(betas[k], 1e-6f);
        rbf[k] = Cc * expf(-b * t * t);
    }

    // Neighbor embedding scatter: aggr[src] += ne_emb[z[dst]] * W
    int zd = z[dn];
#pragma unroll 4
    for (int o = 0; o < 32; ++o) {
        float w = ne_db[o];
#pragma unroll
        for (int i = 0; i < 8; ++i) w += rbf[i] * ne_dw[i * 32 + o];
        w *= C;
        atomicAdd(&aggr[sn * 32 + o], ne_emb[zd * 32 + o] * w);
    }

    // Radial MLP: coeff[l] = silu(rbf@W1+b1) @ W2 + b2
    float coeff[3];
#pragma unroll
    for (int l = 0; l < 3; ++l) {
        float acc = rad_b2[l];
        for (int h = 0; h < 256; ++h) {
            float t = b1s[l * 256 + h];
#pragma unroll
            for (int i = 0; i < 8; ++i) t += rbf[i] * w1s[(l * 8 + i) * 256 + h];
            acc += siluf(t) * w2s[l * 256 + h];
        }
        coeff[l] = acc;
    }

    // eattr = Y * coeff (per L), with nan_to_num
    float e0 = fin0(coeff[0]);
    float e1x = fin0(SQRT3C * hx * coeff[1]);
    float e1y = fin0(SQRT3C * hy * coeff[1]);
    float e1z = fin0(SQRT3C * hz * coeff[1]);
    float y20 = SQRT15C * S2C * 2.0f * hx * hy;
    float y21 = SQRT15C * S2C * 2.0f * hy * hz;
    float y22 = SQRT15C * S6C * (-hx * hx - hy * hy + 2.0f * hz * hz);
    float y23 = SQRT15C * S2C * 2.0f * hx * hz;
    float y24 = SQRT15C * S2C * (hx * hx - hy * hy);
    float e2a = fin0(y20 * coeff[2]);
    float e2b = fin0(y21 * coeff[2]);
    float e2c = fin0(y22 * coeff[2]);
    float e2d = fin0(y23 * coeff[2]);
    float e2e = fin0(y24 * coeff[2]);

    e0g[e] = e0;
    e1g[e * 3 + 0] = e1x;
    e1g[e * 3 + 1] = e1y;
    e1g[e * 3 + 2] = e1z;
    // Qe[i][j] = sum_k e2[k]*B2[k,i,j]
    qeg[e * 9 + 0] = -S6C * e2c + S2C * e2e;
    qeg[e * 9 + 1] = S2C * e2a;
    qeg[e * 9 + 2] = S2C * e2d;
    qeg[e * 9 + 3] = S2C * e2a;
    qeg[e * 9 + 4] = -S6C * e2c - S2C * e2e;
    qeg[e * 9 + 5] = S2C * e2b;
    qeg[e * 9 + 6] = S2C * e2d;
    qeg[e * 9 + 7] = S2C * e2b;
    qeg[e * 9 + 8] = 2.0f * S6C * e2c;
}

// ---------------------------------------------------------------------------
// Stage 2: node init  s = concat(atom_emb[z], aggr) @ ne_cw + ne_cb ; v = 0
// ---------------------------------------------------------------------------
__global__ __launch_bounds__(128)
void pmg_node_init_kernel(const float* __restrict__ atom_emb,
                          const int* __restrict__ z,
                          const float* __restrict__ aggr,
                          const float* __restrict__ ne_cw, // [64,32]
                          const float* __restrict__ ne_cb, // [32]
                          float* __restrict__ sb, float* __restrict__ vb, int N) {
    int n = blockIdx.x * blockDim.x + threadIdx.x;
    if (n >= N) return;
    int zn = z[n];
    float x[64];
#pragma unroll
    for (int i = 0; i < 32; ++i) x[i] = atom_emb[zn * 32 + i];
#pragma unroll
    for (int i = 0; i < 32; ++i) x[32 + i] = aggr[n * 32 + i];
#pragma unroll 4
    for (int o = 0; o < 32; ++o) {
        float t = ne_cb[o];
#pragma unroll
        for (int i = 0; i < 64; ++i) t += x[i] * ne_cw[i * 32 + o];
        sb[n * 32 + o] = t;
    }
#pragma unroll
    for (int k = 0; k < 48; ++k) vb[n * 48 + k] = 0.0f;
}

// ---------------------------------------------------------------------------
// Stage 3: per-layer edge message kernel — WMMA f32 16x16x4.
// One wave32 = one tile of 16 edges.  56 v_wmma per tile.
// ---------------------------------------------------------------------------
__global__ __launch_bounds__(256)
void pmg_layer_edge_kernel(const float* __restrict__ sb,   // [N,32]
                           const float* __restrict__ vb,   // [N,16,3]
                           const float* __restrict__ e0g,
                           const float* __restrict__ e1g,
                           const float* __restrict__ qeg,
                           const float* __restrict__ alpha,
                           const int* __restrict__ srcg,
                           const int* __restrict__ dstg,
                           const float* __restrict__ w00,  // [32,32]
                           const float* __restrict__ w110, // [16,32]
                           const float* __restrict__ w01,  // [32,16]
                           const float* __restrict__ w10,  // [16,16]
                           const float* __restrict__ w12,  // [16,16]
                           float* __restrict__ msb,        // [N,32]
                           float* __restrict__ mvb,        // [N,16,3]
                           int E) {
    int gtid = blockIdx.x * blockDim.x + threadIdx.x;
    int wave = gtid >> 5;
    int lane = threadIdx.x & 31;
    int base = wave * 16;
    if (base >= E) return;

    int row = lane & 15;       // M row (edge) for A frags, N column for B/C frags
    int kh = (lane >> 4) & 1;  // which K pair this lane holds (ISA 16x4 layout)

    int e = base + row;
    if (e >= E) e = E - 1;
    int n = srcg[e];
    const float* srow = sb + (size_t)n * 32;
    const float* vrow = vb + (size_t)n * 48;

    float e0 = e0g[e];
    float e1x = e1g[e * 3 + 0], e1y = e1g[e * 3 + 1], e1z = e1g[e * 3 + 2];
    float q0 = qeg[e * 9 + 0], q1 = qeg[e * 9 + 1], q2 = qeg[e * 9 + 2];
    float q3 = qeg[e * 9 + 3], q4 = qeg[e * 9 + 4], q5 = qeg[e * 9 + 5];
    float q6 = qeg[e * 9 + 6], q7 = qeg[e * 9 + 7], q8 = qeg[e * 9 + 8];

    v8f acc_s0 = {}, acc_s1 = {}, acc_p = {};
    v8f acc_v0 = {}, acc_v1 = {}, acc_v2 = {};

    // ---- Phase 1: K=32 over scalar channels ----
    //   m_s += (ss*e0) @ w00            (two 16-wide N tiles)
    //   p    =  ss     @ w01            (one 16-wide N tile)
#pragma unroll
    for (int kt = 0; kt < 8; ++kt) {
        int k0 = kt * 4 + kh * 2;
        float x0 = srow[k0], x1 = srow[k0 + 1];
        v2f aS;  aS.x = x0;       aS.y = x1;
        v2f a00; a00.x = x0 * e0; a00.y = x1 * e0;
        v2f b0, b1, bp;
        b0.x = w00[k0 * 32 + row];       b0.y = w00[(k0 + 1) * 32 + row];
        b1.x = w00[k0 * 32 + 16 + row];  b1.y = w00[(k0 + 1) * 32 + 16 + row];
        bp.x = w01[k0 * 16 + row];       bp.y = w01[(k0 + 1) * 16 + row];
        acc_s0 = wmma4(a00, b0, acc_s0);
        acc_s1 = wmma4(a00, b1, acc_s1);
        acc_p  = wmma4(aS, bp, acc_p);
    }

    // ---- Phase 2: K=16 over vector channels ----
    //   m_s += (vs.e1) @ w110
    //   m_v[c] += (vs[:,c]*e0) @ w10 + (Qe.vs)[:,c] @ w12
#pragma unroll
    for (int kt = 0; kt < 4; ++kt) {
        int k0 = kt * 4 + kh * 2;
        float va0 = vrow[k0 * 3 + 0], va1 = vrow[k0 * 3 + 1], va2 = vrow[k0 * 3 + 2];
        float vb0 = vrow[(k0 + 1) * 3 + 0], vb1 = vrow[(k0 + 1) * 3 + 1],
              vb2 = vrow[(k0 + 1) * 3 + 2];

        v2f a110;
        a110.x = va0 * e1x + va1 * e1y + va2 * e1z;
        a110.y = vb0 * e1x + vb1 * e1y + vb2 * e1z;
        v2f bs0, bs1;
        bs0.x = w110[k0 * 32 + row];      bs0.y = w110[(k0 + 1) * 32 + row];
        bs1.x = w110[k0 * 32 + 16 + row]; bs1.y = w110[(k0 + 1) * 32 + 16 + row];
        acc_s0 = wmma4(a110, bs0, acc_s0);
        acc_s1 = wmma4(a110, bs1, acc_s1);

        v2f b10, b12;
        b10.x = w10[k0 * 16 + row]; b10.y = w10[(k0 + 1) * 16 + row];
        b12.x = w12[k0 * 16 + row]; b12.y = w12[(k0 + 1) * 16 + row];

        v2f a;
        a.x = va0 * e0; a.y = vb0 * e0; acc_v0 = wmma4(a, b10, acc_v0);
        a.x = va1 * e0; a.y = vb1 * e0; acc_v1 = wmma4(a, b10, acc_v1);
        a.x = va2 * e0; a.y = vb2 * e0; acc_v2 = wmma4(a, b10, acc_v2);

        a.x = q0 * va0 + q1 * va1 + q2 * va2;
        a.y = q0 * vb0 + q1 * vb1 + q2 * vb2;
        acc_v0 = wmma4(a, b12, acc_v0);
        a.x = q3 * va0 + q4 * va1 + q5 * va2;
        a.y = q3 * vb0 + q4 * vb1 + q5 * vb2;
        acc_v1 = wmma4(a, b12, acc_v1);
        a.x = q6 * va0 + q7 * va1 + q8 * va2;
        a.y = q6 * vb0 + q7 * vb1 + q8 * vb2;
        acc_v2 = wmma4(a, b12, acc_v2);
    }

    // ---- Epilogue: + outer(p, e1), *alpha, atomic scatter by dst ----
    int off = (lane < 16) ? 0 : 8;  // C/D layout: vgpr j holds M=j (+8 hi lanes)
#pragma unroll
    for (int j = 0; j < 8; ++j) {
        int em = base + j + off;
        if (em >= E) continue;
        float a_e = alpha[em];
        float m1x = e1g[em * 3 + 0], m1y = e1g[em * 3 + 1], m1z = e1g[em * 3 + 2];
        int dn = dstg[em];
        float pj = acc_p[j];
        atomicAdd(&msb[(size_t)dn * 32 + row], acc_s0[j] * a_e);
        atomicAdd(&msb[(size_t)dn * 32 + 16 + row], acc_s1[j] * a_e);
        atomicAdd(&mvb[(size_t)dn * 48 + row * 3 + 0], (acc_v0[j] + pj * m1x) * a_e);
        atomicAdd(&mvb[(size_t)dn * 48 + row * 3 + 1], (acc_v1[j] + pj * m1y) * a_e);
        atomicAdd(&mvb[(size_t)dn * 48 + row * 3 + 2], (acc_v2[j] + pj * m1z) * a_e);
    }
}

// ---------------------------------------------------------------------------
// Stage 4: per-layer node update (self-mix, gating, LN, gated-norm MLP)
// ---------------------------------------------------------------------------
__global__ __launch_bounds__(128)
void pmg_node_update_kernel(float* __restrict__ sb, float* __restrict__ vb,
                            const float* __restrict__ msb,
                            const float* __restrict__ mvb,
                            const float* __restrict__ self_ws,
                            const float* __restrict__ self_wv,
                            const float* __restrict__ pre_ws,  // [32,48]
                            const float* __restrict__ pre_wv,
                            const float* __restrict__ post_ws,
                            const float* __restrict__ post_wv,
                            const float* __restrict__ ln_g,
                            const float* __restrict__ ln_b,
                            const float* __restrict__ g_w1,
                            const float* __restrict__ g_b1,
                            const float* __restrict__ g_w2,
                            const float* __restrict__ g_b2, int N) {
    int n = blockIdx.x * blockDim.x + threadIdx.x;
    if (n >= N) return;
    float s0[32], v0[48], s1[32], v1[48], ps[48];
#pragma unroll
    for (int i = 0; i < 32; ++i) s0[i] = sb[n * 32 + i];
#pragma unroll
    for (int k = 0; k < 48; ++k) v0[k] = vb[n * 48 + k];

    // self interaction + messages
#pragma unroll 4
    for (int o = 0; o < 32; ++o) {
        float t = msb[n * 32 + o];
#pragma unroll
        for (int i = 0; i < 32; ++i) t += s0[i] * self_ws[i * 32 + o];
        s1[o] = t;
    }
#pragma unroll 2
    for (int o = 0; o < 16; ++o)
#pragma unroll
        for (int c = 0; c < 3; ++c) {
            float t = mvb[n * 48 + o * 3 + c];
#pragma unroll
            for (int i = 0; i < 16; ++i) t += v0[i * 3 + c] * self_wv[i * 16 + o];
            v1[o * 3 + c] = t;
        }
    // pre
#pragma unroll 4
    for (int o = 0; o < 48; ++o) {
        float t = 0.0f;
#pragma unroll
        for (int i = 0; i < 32; ++i) t += s1[i] * pre_ws[i * 48 + o];
        ps[o] = t;
    }
#pragma unroll 2
    for (int o = 0; o < 16; ++o)
#pragma unroll
        for (int c = 0; c < 3; ++c) {
            float t = 0.0f;
#pragma unroll
            for (int i = 0; i < 16; ++i) t += v1[i * 3 + c] * pre_wv[i * 16 + o];
            v0[o * 3 + c] = t;  // pv
        }
    // gate
#pragma unroll
    for (int o = 0; o < 32; ++o) s0[o] = siluf(ps[o]);  // sc
#pragma unroll
    for (int o = 0; o < 16; ++o) {
        float g = sigf(ps[32 + o]);
#pragma unroll
        for (int c = 0; c < 3; ++c) v1[o * 3 + c] = v0[o * 3 + c] * g;  // gated v
    }
    // post
#pragma unroll 4
    for (int o = 0; o < 32; ++o) {
        float t = 0.0f;
#pragma unroll
        for (int i = 0; i < 32; ++i) t += s0[i] * post_ws[i * 32 + o];
        s1[o] = t;
    }
#pragma unroll 2
    for (int o = 0; o < 16; ++o)
#pragma unroll
        for (int c = 0; c < 3; ++c) {
            float t = 0.0f;
#pragma unroll
            for (int i = 0; i < 16; ++i) t += v1[i * 3 + c] * post_wv[i * 16 + o];
            v0[o * 3 + c] = t;  // v2
        }
    // layernorm on s1
    float mu = 0.0f;
#pragma unroll
    for (int o = 0; o < 32; ++o) mu += s1[o];
    mu *= (1.0f / 32.0f);
    float var = 0.0f;
#pragma unroll
    for (int o = 0; o < 32; ++o) {
        float dlt = s1[o] - mu;
        var += dlt * dlt;
    }
    var *= (1.0f / 32.0f);
    float rs = rsqrtf(var + 1e-5f);
#pragma unroll
    for (int o = 0; o < 32; ++o) s1[o] = (s1[o] - mu) * rs * ln_g[o] + ln_b[o];
    // gated-norm MLP
    float vn16[16], hid[16];
#pragma unroll
    for (int o = 0; o < 16; ++o) {
        float t = v0[o * 3 + 0] * v0[o * 3 + 0] + v0[o * 3 + 1] * v0[o * 3 + 1] +
                  v0[o * 3 + 2] * v0[o * 3 + 2];
        vn16[o] = sqrtf(t + 1e-12f);
    }
#pragma unroll
    for (int o = 0; o < 16; ++o) {
        float t = g_b1[o];
#pragma unroll
        for (int i = 0; i < 16; ++i) t += vn16[i] * g_w1[i * 16 + o];
        hid[o] = siluf(t);
    }
#pragma unroll 4
    for (int o = 0; o < 32; ++o) {
        float t = g_b2[o];
#pragma unroll
        for (int i = 0; i < 16; ++i) t += hid[i] * g_w2[i * 32 + o];
        sb[n * 32 + o] = s1[o] + t;
    }
#pragma unroll
    for (int k = 0; k < 48; ++k) vb[n * 48 + k] = v0[k];
}

// ---------------------------------------------------------------------------
// Stage 5: output  out = concat(s, v.reshape(N,48))
// ---------------------------------------------------------------------------
__global__ __launch_bounds__(256)
void pmg_write_out_kernel(const float* __restrict__ sb, const float* __restrict__ vb,
                          float* __restrict__ out, int N) {
    int i = blockIdx.x * blockDim.x + threadIdx.x;
    int total = N * 80;
    if (i >= total) return;
    int n = i / 80, k = i % 80;
    out[i] = (k < 32) ? sb[n * 32 + k] : vb[n * 48 + (k - 32)];
}

// ---------------------------------------------------------------------------
extern "C" void kernel_launch(void* const* d_in, const int* in_sizes, int n_in,
                              void* d_out, int out_size, void* d_ws, size_t ws_size,
                              hipStream_t stream) {
    const float* pos      = (const float*)d_in[0];
    const float* alpha    = (const float*)d_in[1];
    const float* atom_emb = (const float*)d_in[2];
    const float* ne_emb   = (const float*)d_in[3];
    const float* ne_dw    = (const float*)d_in[4];
    const float* ne_db    = (const float*)d_in[5];
    const float* ne_cw    = (const float*)d_in[6];
    const float* ne_cb    = (const float*)d_in[7];
    const float* means    = (const float*)d_in[8];
    const float* betas    = (const float*)d_in[9];
    const float* rad_w1   = (const float*)d_in[10];
    const float* rad_b1   = (const float*)d_in[11];
    const float* rad_w2   = (const float*)d_in[12];
    const float* rad_b2   = (const float*)d_in[13];
    const float* tp_w00   = (const float*)d_in[14];
    const float* tp_w110  = (const float*)d_in[15];
    const float* tp_w01   = (const float*)d_in[16];
    const float* tp_w10   = (const float*)d_in[17];
    const float* tp_w12   = (const float*)d_in[18];
    const float* self_ws  = (const float*)d_in[19];
    const float* self_wv  = (const float*)d_in[20];
    const float* pre_ws   = (const float*)d_in[21];
    const float* pre_wv   = (const float*)d_in[22];
    const float* post_ws  = (const float*)d_in[23];
    const float* post_wv  = (const float*)d_in[24];
    const float* ln_g     = (const float*)d_in[25];
    const float* ln_b     = (const float*)d_in[26];
    const float* g_w1     = (const float*)d_in[27];
    const float* g_b1     = (const float*)d_in[28];
    const float* g_w2     = (const float*)d_in[29];
    const float* g_b2     = (const float*)d_in[30];
    const int* z          = (const int*)d_in[31];
    const int* ei         = (const int*)d_in[32];

    const int N = in_sizes[0] / 3;
    const int E = in_sizes[1];

    float* w = (float*)d_ws;
    float* e0g  = w;
    float* e1g  = e0g + (size_t)E;
    float* qeg  = e1g + (size_t)3 * E;
    float* aggr = qeg + (size_t)9 * E;
    float* sbuf = aggr + (size_t)32 * N;
    float* vbuf = sbuf + (size_t)32 * N;
    float* msb  = vbuf + (size_t)48 * N;
    float* mvb  = msb + (size_t)32 * N;  // ms+mv contiguous: 80N floats

    hipMemsetAsync(aggr, 0, (size_t)32 * N * sizeof(float), stream);

    pmg_edge_pre_kernel<<<(E + 255) / 256, 256, 0, stream>>>(
        pos, ei, means, betas, ne_dw, ne_db, ne_emb, z, rad_w1, rad_b1, rad_w2,
        rad_b2, e0g, e1g, qeg, aggr, E);

    pmg_node_init_kernel<<<(N + 127) / 128, 128, 0, stream>>>(
        atom_emb, z, aggr, ne_cw, ne_cb, sbuf, vbuf, N);

    const int tiles = (E + 15) / 16;
    const int blocks = (tiles + 7) / 8;  // 8 waves (wave32) per 256-thread block

    for (int li = 0; li < 4; ++li) {
        hipMemsetAsync(msb, 0, (size_t)80 * N * sizeof(float), stream);
        pmg_layer_edge_kernel<<<blocks, 256, 0, stream>>>(
            sbuf, vbuf, e0g, e1g, qeg, alpha, ei, ei + E,
            tp_w00 + li * 32 * 32, tp_w110 + li * 16 * 32, tp_w01 + li * 32 * 16,
            tp_w10 + li * 16 * 16, tp_w12 + li * 16 * 16, msb, mvb, E);
        pmg_node_update_kernel<<<(N + 127) / 128, 128, 0, stream>>>(
            sbuf, vbuf, msb, mvb, self_ws + li * 32 * 32, self_wv + li * 16 * 16,
            pre_ws + li * 32 * 48, pre_wv + li * 16 * 16, post_ws + li * 32 * 32,
            post_wv + li * 16 * 16, ln_g + li * 32, ln_b + li * 32,
            g_w1 + li * 16 * 16, g_b1 + li * 16, g_w2 + li * 16 * 32,
            g_b2 + li * 32, N);
    }

    pmg_write_out_kernel<<<(N * 80 + 255) / 256, 256, 0, stream>>>(
        sbuf, vbuf, (float*)d_out, N);
}